// BiConv_40415642256025
// MI455X (gfx1250) — compile-verified
//
#include <hip/hip_runtime.h>

// BiConv: out = relu(norm*(x + scatter(x[src]->dst)) @ Wout^T)
//             + relu(norm_t*(x + scatter(x[dst]->src)) @ Wback^T)
// Phase 1: init agg buffers with x (float4 copies)
// Phase 2: edge scatter with hardware global_atomic_add_f32 (L2-resident)
// Phase 3: fused dual-GEMM epilogue using V_WMMA_F32_16X16X4_F32

typedef __attribute__((ext_vector_type(2))) float v2f;
typedef __attribute__((ext_vector_type(8))) float v8f;

#define CH 64

__global__ void __launch_bounds__(256)
init_agg_kernel(const float* __restrict__ x,
                float* __restrict__ agg_out,
                float* __restrict__ agg_back,
                long n4) {
    long i = (long)blockIdx.x * blockDim.x + threadIdx.x;
    if (i >= n4) return;
    float4 v = ((const float4*)x)[i];
    ((float4*)agg_out)[i]  = v;
    ((float4*)agg_back)[i] = v;
}

__global__ void __launch_bounds__(256)
edge_scatter_kernel(const float* __restrict__ x,
                    const int* __restrict__ src,
                    const int* __restrict__ dst,
                    float* __restrict__ agg_out,
                    float* __restrict__ agg_back,
                    int nedges) {
    long t = (long)blockIdx.x * blockDim.x + threadIdx.x;
    int e = (int)(t >> 4);              // 16 threads per edge
    if (e >= nedges) return;
    int c4 = ((int)t & 15) << 2;        // 4 channels per thread

    int s = src[e];
    int d = dst[e];

    float4 xs = *(const float4*)(x + (size_t)s * CH + c4);  // x[src] -> dst
    float4 xd = *(const float4*)(x + (size_t)d * CH + c4);  // x[dst] -> src

    float* ao = agg_out  + (size_t)d * CH + c4;
    float* ab = agg_back + (size_t)s * CH + c4;

    // Non-returning hardware float atomics (relaxed, device scope)
    __hip_atomic_fetch_add(ao + 0, xs.x, __ATOMIC_RELAXED, __HIP_MEMORY_SCOPE_AGENT);
    __hip_atomic_fetch_add(ao + 1, xs.y, __ATOMIC_RELAXED, __HIP_MEMORY_SCOPE_AGENT);
    __hip_atomic_fetch_add(ao + 2, xs.z, __ATOMIC_RELAXED, __HIP_MEMORY_SCOPE_AGENT);
    __hip_atomic_fetch_add(ao + 3, xs.w, __ATOMIC_RELAXED, __HIP_MEMORY_SCOPE_AGENT);
    __hip_atomic_fetch_add(ab + 0, xd.x, __ATOMIC_RELAXED, __HIP_MEMORY_SCOPE_AGENT);
    __hip_atomic_fetch_add(ab + 1, xd.y, __ATOMIC_RELAXED, __HIP_MEMORY_SCOPE_AGENT);
    __hip_atomic_fetch_add(ab + 2, xd.z, __ATOMIC_RELAXED, __HIP_MEMORY_SCOPE_AGENT);
    __hip_atomic_fetch_add(ab + 3, xd.w, __ATOMIC_RELAXED, __HIP_MEMORY_SCOPE_AGENT);
}

// One block = 4 waves = one 16-node tile; wave w handles output channels [16w, 16w+16).
// Per wave: D(16x16) = A(16x64) x B(64x16) twice (Wout / Wback branches),
// K-loop of 16 x V_WMMA_F32_16X16X4_F32 per branch, interleaved for ILP.
__global__ void __launch_bounds__(128)
gemm_fused_kernel(const float* __restrict__ agg_out,
                  const float* __restrict__ agg_back,
                  const float* __restrict__ norm,
                  const float* __restrict__ norm_t,
                  const float* __restrict__ W_out,
                  const float* __restrict__ W_back,
                  float* __restrict__ out) {
    const int lane = threadIdx.x & 31;
    const int wave = threadIdx.x >> 5;   // n-block 0..3
    const int m  = lane & 15;            // row (A) / column N (B,C,D)
    const int hi = lane >> 4;            // lane half selects K pair / M half
    const int base = blockIdx.x * 16;
    const int node = base + m;

    // B fragments: B[k][n] = W[n][k], n = 16*wave + m, k = 4*kk + 2*hi + {0,1}
    const float* wo = W_out  + (size_t)(wave * 16 + m) * CH + 2 * hi;
    const float* wb = W_back + (size_t)(wave * 16 + m) * CH + 2 * hi;
    v2f bo[16], bb[16];
#pragma unroll
    for (int kk = 0; kk < 16; ++kk) {
        bo[kk] = *(const v2f*)(wo + 4 * kk);
        bb[kk] = *(const v2f*)(wb + 4 * kk);
    }

    const float sO = norm[node];
    const float sB = norm_t[node];

    // A fragments: A[m][k], k = 4*kk + 2*hi + {0,1}; norm folded into A
    const float* ao = agg_out  + (size_t)node * CH + 2 * hi;
    const float* ab = agg_back + (size_t)node * CH + 2 * hi;

    v8f acc1 = {0.f, 0.f, 0.f, 0.f, 0.f, 0.f, 0.f, 0.f};
    v8f acc2 = {0.f, 0.f, 0.f, 0.f, 0.f, 0.f, 0.f, 0.f};
#pragma unroll
    for (int kk = 0; kk < 16; ++kk) {
        v2f a1 = *(const v2f*)(ao + 4 * kk);
        v2f a2 = *(const v2f*)(ab + 4 * kk);
        a1 *= sO;
        a2 *= sB;
        acc1 = __builtin_amdgcn_wmma_f32_16x16x4_f32(
            false, a1, false, bo[kk], (short)0, acc1, false, false);
        acc2 = __builtin_amdgcn_wmma_f32_16x16x4_f32(
            false, a2, false, bb[kk], (short)0, acc2, false, false);
    }

    // D element (M = v + 8*hi, N = m); out[node_row][16*wave + m]
    float* o = out + (size_t)base * CH + wave * 16 + m;
#pragma unroll
    for (int v = 0; v < 8; ++v) {
        float r1 = acc1[v] > 0.f ? acc1[v] : 0.f;
        float r2 = acc2[v] > 0.f ? acc2[v] : 0.f;
        o[(size_t)(v + 8 * hi) * CH] = r1 + r2;
    }
}

extern "C" void kernel_launch(void* const* d_in, const int* in_sizes, int n_in,
                              void* d_out, int out_size, void* d_ws, size_t ws_size,
                              hipStream_t stream) {
    (void)n_in; (void)out_size; (void)ws_size;

    const float* x       = (const float*)d_in[0];
    const int*   sources = (const int*)d_in[1];
    const int*   targets = (const int*)d_in[2];
    const float* norm    = (const float*)d_in[3];
    const float* norm_t  = (const float*)d_in[4];
    const float* W_out   = (const float*)d_in[5];
    const float* W_back  = (const float*)d_in[6];
    float*       out     = (float*)d_out;

    const int n = in_sizes[0] / CH;    // 200000
    const int E = in_sizes[1];         // 3200000

    float* agg_out  = (float*)d_ws;                       // N*64 floats
    float* agg_back = agg_out + (size_t)n * CH;           // N*64 floats

    // Phase 1: agg_* = x
    long n4 = (long)n * CH / 4;
    init_agg_kernel<<<(unsigned)((n4 + 255) / 256), 256, 0, stream>>>(
        x, agg_out, agg_back, n4);

    // Phase 2: scatter both directions (L2-resident atomics)
    long tE = (long)E * 16;
    edge_scatter_kernel<<<(unsigned)((tE + 255) / 256), 256, 0, stream>>>(
        x, sources, targets, agg_out, agg_back, E);

    // Phase 3: fused norm-scale + dual GEMM + relu + add (N divisible by 16)
    gemm_fused_kernel<<<(unsigned)(n / 16), 128, 0, stream>>>(
        agg_out, agg_back, norm, norm_t, W_out, W_back, out);
}